// DetSegSampling_34076270527052
// MI455X (gfx1250) — compile-verified
//
#include <hip/hip_runtime.h>
#include <hip/hip_bf16.h>

#define BB   2
#define QQ   900
#define NN   6
#define GG   4
#define PP   8
#define LL   4
#define EMBD 256
#define CC   64
#define NOUT 224              // 96 (off) + 128 (sw logits)
#define MROWS (BB * QQ)       // 1800
#define MTILES ((MROWS + 15) / 16)   // 113
#define NTILES (NOUT / 16)           // 14
#define IMG_W 704.0f
#define IMG_H 256.0f
#define EPS_F 1e-5f

typedef float v2f __attribute__((ext_vector_type(2)));
typedef float v8f __attribute__((ext_vector_type(8)));

// ---------------------------------------------------------------------------
// Kernel 1: OFF_SW = query @ [W_off | W_sw] + [b_off | b_sw]
// One wave per 16x16 output tile, K swept in steps of 4 with
// V_WMMA_F32_16X16X4_F32 (full fp32 matrix engine; result feeds coordinates,
// so fp32 precision is the right choice and the GEMM is tiny anyway).
//
// Out-of-range M rows: clamp the source row instead of zeroing — a wrong A
// row m only corrupts D row m, and those rows are masked at the store. This
// keeps the inner loop completely branch-free (no exec-mask juggling):
//   1x global_load_b64 (A float2, 8B aligned) + 2x global_load_b32 (B) + WMMA.
//
// A frag (16x4 f32): lanes 0-15 -> K={k,k+1}, lanes 16-31 -> K={k+2,k+3}
// B frag (4x16 f32): mirrored (row-of-B striped across lanes)
// C/D (16x16 f32):   VGPR v, lane l -> row = v + 8*(l>=16), col = l&15
// ---------------------------------------------------------------------------
__global__ void __launch_bounds__(32)
gemm_offsw_kernel(const float* __restrict__ query,
                  const float* __restrict__ Woff,
                  const float* __restrict__ boff,
                  const float* __restrict__ Wsw,
                  const float* __restrict__ bsw,
                  float* __restrict__ outws)
{
    const int tile = blockIdx.x;
    const int mt = tile / NTILES;
    const int nt = tile % NTILES;
    const int lane = threadIdx.x & 31;
    const int half = lane >> 4;           // 0 or 1
    const int mrow = mt * 16 + (lane & 15);
    const int ncol = nt * 16 + (lane & 15);

    // Select weight/bias source for this lane's output column
    const float* Bsrc;
    int bstride, bcol;
    float bias;
    if (ncol < 96) { Bsrc = Woff; bstride = 96;  bcol = ncol;      bias = boff[bcol]; }
    else           { Bsrc = Wsw;  bstride = 128; bcol = ncol - 96; bias = bsw[bcol]; }

    // Clamp OOB rows (their results are masked at store time).
    const int mrow_c = (mrow < MROWS) ? mrow : (MROWS - 1);
    const int koff = half * 2;
    const float* Aptr = query + (size_t)mrow_c * EMBD + koff;   // 8B aligned
    const float* Bptr = Bsrc + (size_t)koff * bstride + bcol;

    v8f c = {0.f, 0.f, 0.f, 0.f, 0.f, 0.f, 0.f, 0.f};

    #pragma unroll 4
    for (int k = 0; k < EMBD; k += 4) {
        const v2f a = *(const v2f*)(Aptr + k);
        v2f bf;
        bf.x = Bptr[(size_t)k * bstride];
        bf.y = Bptr[(size_t)(k + 1) * bstride];
        c = __builtin_amdgcn_wmma_f32_16x16x4_f32(false, a, false, bf,
                                                  (short)0, c, false, false);
    }

    #pragma unroll
    for (int v = 0; v < 8; ++v) {
        const int row = mt * 16 + v + 8 * half;
        if (row < MROWS) {
            outws[(size_t)row * NOUT + ncol] = c[v] + bias;
        }
    }
}

// ---------------------------------------------------------------------------
// Kernel 2: projection + softmax + masked multi-level bilinear gather.
// Grid (G, Q, B), block 256 = 8 waves. Wave = sample point p; lanes = channels
// (c=lane, c=lane+32) so camera-validity and tap-bounds branches are
// wave-uniform and invalid cameras are skipped entirely. Feature maps (46 MB)
// are L2-resident (192 MB); output stores are fully coalesced.
// ---------------------------------------------------------------------------
__global__ void __launch_bounds__(256)
sample_kernel(const float* __restrict__ refp,
              const float* __restrict__ l2i,
              const float* __restrict__ ws,
              const float* __restrict__ f0,
              const float* __restrict__ f1,
              const float* __restrict__ f2,
              const float* __restrict__ f3,
              float* __restrict__ out)
{
    const int g = blockIdx.x;
    const int q = blockIdx.y;
    const int b = blockIdx.z;
    const int p = threadIdx.x >> 5;
    const int lane = threadIdx.x & 31;
    const int bq = b * QQ + q;
    const int gp = g * PP + p;

    const float* row = ws + (size_t)bq * NOUT;

    // offsets for this (g,p)
    const float ox = row[gp * 3 + 0];
    const float oy = row[gp * 3 + 1];
    const float oz = row[gp * 3 + 2];

    // softmax over L=4 levels
    const float* lgp = row + 96 + gp * 4;
    float l0 = lgp[0], l1 = lgp[1], l2 = lgp[2], l3 = lgp[3];
    float mx = fmaxf(fmaxf(l0, l1), fmaxf(l2, l3));
    float e0 = expf(l0 - mx), e1 = expf(l1 - mx);
    float e2 = expf(l2 - mx), e3 = expf(l3 - mx);
    float inv = 1.0f / (e0 + e1 + e2 + e3);
    float sw[4] = {e0 * inv, e1 * inv, e2 * inv, e3 * inv};

    // world-space point
    const float rx = refp[bq * 3 + 0];
    const float ry = refp[bq * 3 + 1];
    const float rz = refp[bq * 3 + 2];
    const float px = rx * 102.4f - 51.2f + ox;
    const float py = ry * 102.4f - 51.2f + oy;
    const float pz = rz * 8.0f   - 5.0f  + oz;

    const int   Hs[4]  = {32, 16, 8, 4};
    const int   Wds[4] = {88, 44, 22, 11};
    const int   HWs[4] = {2816, 704, 176, 44};
    const float* feats[4] = {f0, f1, f2, f3};

    const int c0 = lane;
    const int c1 = lane + 32;
    float acc0 = 0.f, acc1 = 0.f;

    for (int n = 0; n < NN; ++n) {
        const float* M = l2i + (size_t)(b * NN + n) * 16;
        const float cam0 = M[0] * px + M[1] * py + M[2]  * pz + M[3];
        const float cam1 = M[4] * px + M[5] * py + M[6]  * pz + M[7];
        const float cam2 = M[8] * px + M[9] * py + M[10] * pz + M[11];
        const float d = fmaxf(cam2, EPS_F);
        const float u = cam0 / d / IMG_W;
        const float v = cam1 / d / IMG_H;
        const bool valid = (cam2 > EPS_F) & (u > 0.f) & (u < 1.f)
                                         & (v > 0.f) & (v < 1.f);
        if (!valid) continue;   // wave-uniform: all lanes share (b,q,g,p,n)

        #pragma unroll
        for (int l = 0; l < LL; ++l) {
            const int   Hl = Hs[l], Wl = Wds[l], HWl = HWs[l];
            const float x = u * (float)Wl - 0.5f;
            const float y = v * (float)Hl - 0.5f;
            const float x0f = floorf(x);
            const float y0f = floorf(y);
            const float wx1 = x - x0f;
            const float wy1 = y - y0f;
            const int x0 = (int)x0f;
            const int y0 = (int)y0f;
            const bool bx0 = (x0 >= 0)     & (x0 < Wl);
            const bool bx1 = (x0 + 1 >= 0) & (x0 + 1 < Wl);
            const bool by0 = (y0 >= 0)     & (y0 < Hl);
            const bool by1 = (y0 + 1 >= 0) & (y0 + 1 < Hl);

            const float w00 = (1.f - wx1) * (1.f - wy1);
            const float w10 = wx1 * (1.f - wy1);
            const float w01 = (1.f - wx1) * wy1;
            const float w11 = wx1 * wy1;

            const float* base = feats[l]
                + (size_t)((b * NN + n) * (GG * CC) + g * CC) * HWl;
            const float* ch0 = base + (size_t)c0 * HWl;
            const float* ch1 = base + (size_t)c1 * HWl;

            const int i00 = y0 * Wl + x0;
            const int i10 = i00 + 1;
            const int i01 = i00 + Wl;
            const int i11 = i01 + 1;

            float s0 = 0.f, s1 = 0.f;
            if (bx0 & by0) { s0 += w00 * ch0[i00]; s1 += w00 * ch1[i00]; }
            if (bx1 & by0) { s0 += w10 * ch0[i10]; s1 += w10 * ch1[i10]; }
            if (bx0 & by1) { s0 += w01 * ch0[i01]; s1 += w01 * ch1[i01]; }
            if (bx1 & by1) { s0 += w11 * ch0[i11]; s1 += w11 * ch1[i11]; }

            acc0 += sw[l] * s0;
            acc1 += sw[l] * s1;
        }
    }

    const size_t obase = ((((size_t)b * QQ + q) * GG + g) * PP + p) * CC;
    out[obase + c0] = acc0;
    out[obase + c1] = acc1;
}

// ---------------------------------------------------------------------------
extern "C" void kernel_launch(void* const* d_in, const int* in_sizes, int n_in,
                              void* d_out, int out_size, void* d_ws, size_t ws_size,
                              hipStream_t stream) {
    const float* query = (const float*)d_in[0];
    const float* refp  = (const float*)d_in[1];
    const float* l2i   = (const float*)d_in[2];
    const float* Woff  = (const float*)d_in[3];
    const float* boff  = (const float*)d_in[4];
    const float* Wsw   = (const float*)d_in[5];
    const float* bsw   = (const float*)d_in[6];
    const float* f0    = (const float*)d_in[7];
    const float* f1    = (const float*)d_in[8];
    const float* f2    = (const float*)d_in[9];
    const float* f3    = (const float*)d_in[10];

    float* ws  = (float*)d_ws;     // 1800 x 224 f32 = ~1.6 MB
    float* out = (float*)d_out;

    gemm_offsw_kernel<<<MTILES * NTILES, 32, 0, stream>>>(
        query, Woff, boff, Wsw, bsw, ws);

    dim3 grid(GG, QQ, BB);
    sample_kernel<<<grid, 256, 0, stream>>>(
        refp, l2i, ws, f0, f1, f2, f3, out);
}